// CSNN_45337674776868
// MI455X (gfx1250) — compile-verified
//
#include <hip/hip_runtime.h>

typedef __attribute__((ext_vector_type(2))) float v2f;
typedef __attribute__((ext_vector_type(4))) float v4f;
typedef __attribute__((ext_vector_type(8))) float v8f;

#define T_STEPS 50
#define BATCH   128
#define AXON    1000
#define NEURON  10000
#define BETA    0.95f
#define THRESH  1.0f

// ---------------------------------------------------------------------------
// Kernel A: cur[m,n] = sum_k x[m,k] * W[n,k] + b[n]   (M=128, N=10000, K=1000)
//
// Register-blocked WMMA GEMM: each wave owns a 64x32 C block = 4 M-tiles x
// 2 N-tiles = 8 v8f accumulators. Per K=4 step: 4 A frags + 2 B frags
// (6 x global_load_b64) feed 8 v_wmma_f32_16x16x4_f32.
// N = 10000 = 312 full 32-col pairs + 16-col remainder: remainder handled by
// wave-uniform clamp on loads (keeps EXEC all-1s for WMMA) + store guard.
// ---------------------------------------------------------------------------
__global__ __launch_bounds__(256) void snn_gemm_wmma(
    const float* __restrict__ x,    // [128, 1000]
    const float* __restrict__ W,    // [10000, 1000]
    const float* __restrict__ b,    // [10000]
    float* __restrict__ cur)        // [128, 10000]
{
    const int lane = threadIdx.x & 31;
    const int wave = threadIdx.x >> 5;
    const int gw   = blockIdx.x * 8 + wave;        // global wave id

    const int NTILES = NEURON / 16;                // 625
    const int NPAIR  = (NTILES + 1) / 2;           // 313
    if (gw >= 2 * NPAIR) return;                   // wave-uniform exit

    const int m0 = (gw & 1) * 64;                  // M block: 0 or 64
    const int n0 = (gw >> 1) * 32;                 // first N tile
    const int n1 = n0 + 16;                        // second N tile
    const bool n1ok = (n1 < NEURON);
    const int  n1c  = n1ok ? n1 : n0;              // clamp for loads (uniform)

    const int row = lane & 15;                     // A row / B col / C col
    const int hi  = lane >> 4;                     // lane half select

    // A 16x4 f32 frag: lane holds {A[row][2*hi], A[row][2*hi+1]} (contig pair)
    // B 4x16 f32 frag: lane holds {W[n+row][2*hi], W[n+row][2*hi+1]}
    const float* xa  = x + (size_t)(m0 + row) * AXON + 2 * hi;
    const float* wb0 = W + (size_t)(n0  + row) * AXON + 2 * hi;
    const float* wb1 = W + (size_t)(n1c + row) * AXON + 2 * hi;

    // C layout: VGPR r, lanes 0-15 -> (M=r, N=lane); lanes 16-31 -> (M=r+8).
    // Bias is per-column -> broadcast into every accumulator register.
    const float bias0 = b[n0 + row];
    const float bias1 = b[n1c + row];
    v8f c[8];                                       // c[2*i+j]: M-tile i, N-tile j
#pragma unroll
    for (int i = 0; i < 4; ++i)
#pragma unroll
        for (int r = 0; r < 8; ++r) { c[2*i][r] = bias0; c[2*i+1][r] = bias1; }

    for (int k = 0; k < AXON; k += 4) {
        v2f a0 = *(const v2f*)(xa + k);
        v2f a1 = *(const v2f*)(xa + 16 * AXON + k);
        v2f a2 = *(const v2f*)(xa + 32 * AXON + k);
        v2f a3 = *(const v2f*)(xa + 48 * AXON + k);
        v2f b0 = *(const v2f*)(wb0 + k);
        v2f b1 = *(const v2f*)(wb1 + k);
        c[0] = __builtin_amdgcn_wmma_f32_16x16x4_f32(false, a0, false, b0, (short)0, c[0], false, false);
        c[1] = __builtin_amdgcn_wmma_f32_16x16x4_f32(false, a0, false, b1, (short)0, c[1], false, false);
        c[2] = __builtin_amdgcn_wmma_f32_16x16x4_f32(false, a1, false, b0, (short)0, c[2], false, false);
        c[3] = __builtin_amdgcn_wmma_f32_16x16x4_f32(false, a1, false, b1, (short)0, c[3], false, false);
        c[4] = __builtin_amdgcn_wmma_f32_16x16x4_f32(false, a2, false, b0, (short)0, c[4], false, false);
        c[5] = __builtin_amdgcn_wmma_f32_16x16x4_f32(false, a2, false, b1, (short)0, c[5], false, false);
        c[6] = __builtin_amdgcn_wmma_f32_16x16x4_f32(false, a3, false, b0, (short)0, c[6], false, false);
        c[7] = __builtin_amdgcn_wmma_f32_16x16x4_f32(false, a3, false, b1, (short)0, c[7], false, false);
    }

#pragma unroll
    for (int i = 0; i < 4; ++i) {
        const int mrow = m0 + i * 16 + 8 * hi;
#pragma unroll
        for (int r = 0; r < 8; ++r) {
            cur[(size_t)(mrow + r) * NEURON + n0 + row] = c[2*i][r];
            if (n1ok)
                cur[(size_t)(mrow + r) * NEURON + n1 + row] = c[2*i+1][r];
        }
    }
}

// ---------------------------------------------------------------------------
// Kernel B: LIF recurrence, T=50 steps, reset-by-subtraction (snntorch Leaky).
// One thread owns 4 consecutive neurons; cur & mem live in registers; the
// 512 MB of spk/mem output is streamed with coalesced non-temporal b128
// stores (keep L2 for cur/W, don't pollute it with write-once output).
// Output: d_out = spk_rec[50][128][10000] ++ mem_rec[50][128][10000].
// ---------------------------------------------------------------------------
__global__ __launch_bounds__(256) void snn_lif(
    const float* __restrict__ cur,  // [128, 10000]
    float* __restrict__ out)
{
    const int idx    = blockIdx.x * blockDim.x + threadIdx.x;   // float4 index
    const int total4 = BATCH * NEURON / 4;
    if (idx >= total4) return;

    const v4f c = ((const v4f*)cur)[idx];
    v4f mem = {0.0f, 0.0f, 0.0f, 0.0f};

    float* spk_base = out;
    float* mem_base = out + (size_t)T_STEPS * BATCH * NEURON;
    const size_t elem = (size_t)idx * 4;

    for (int t = 0; t < T_STEPS; ++t) {
        v4f spk;
#pragma unroll
        for (int j = 0; j < 4; ++j) {
            // reset computed from PREVIOUS mem (reset='subtract')
            const float reset = (mem[j] > THRESH) ? THRESH : 0.0f;
            mem[j] = fmaf(BETA, mem[j], c[j]) - reset;
            spk[j] = (mem[j] > THRESH) ? 1.0f : 0.0f;
        }
        const size_t off = (size_t)t * (BATCH * NEURON) + elem;
        __builtin_nontemporal_store(spk, (v4f*)(spk_base + off));
        __builtin_nontemporal_store(mem, (v4f*)(mem_base + off));
    }
}

// ---------------------------------------------------------------------------
extern "C" void kernel_launch(void* const* d_in, const int* in_sizes, int n_in,
                              void* d_out, int out_size, void* d_ws, size_t ws_size,
                              hipStream_t stream) {
    (void)in_sizes; (void)n_in; (void)out_size; (void)ws_size;
    const float* x = (const float*)d_in[0];
    const float* W = (const float*)d_in[1];
    const float* b = (const float*)d_in[2];
    float* cur = (float*)d_ws;          // 128*10000*4 = 5.12 MB scratch
    float* out = (float*)d_out;

    // 626 working waves (2 M-blocks x 313 N-pairs), 8 waves per block.
    const int nwaves = 2 * ((NEURON / 16 + 1) / 2);
    snn_gemm_wmma<<<dim3((nwaves + 7) / 8), dim3(256), 0, stream>>>(x, W, b, cur);

    const int total4 = BATCH * NEURON / 4;
    snn_lif<<<dim3((total4 + 255) / 256), dim3(256), 0, stream>>>(cur, out);
}